// GraphClassifier_21028159881816
// MI455X (gfx1250) — compile-verified
//
#include <hip/hip_runtime.h>
#include <hip/hip_bf16.h>

// ---------------------------------------------------------------------------
// Problem constants (match the reference)
// ---------------------------------------------------------------------------
#define NND   32000      // nodes
#define HD    128        // hidden dim
#define LD    3          // layers
#define RD    8          // relations
#define ED    512000     // edges
#define BSD   32         // batch (subgraphs)
#define KCL   64         // clusters
#define KREL  (RD * HD)  // 1024 : flattened (r,i) K-dim of the relation GEMM
#define KTOT  (KREL + HD)// 1152 : concat [Wrel ; Wself] K-dim

typedef __attribute__((ext_vector_type(2))) float v2f;
typedef __attribute__((ext_vector_type(8))) float v8f;

// ---------------------------------------------------------------------------
// 0) Build transposed concat weights, once per launch:
//    WT[l][o][k] = (k < 1024) ? W_rel[l][k][o] : W_self[l][k-1024][o]
//    Makes the WMMA B-fragment loads contiguous (b64) like the A side.
// ---------------------------------------------------------------------------
__global__ void build_wt(const float* __restrict__ W_rel,
                         const float* __restrict__ W_self,
                         float* __restrict__ WT) {
    size_t idx = (size_t)blockIdx.x * blockDim.x + threadIdx.x;
    size_t total = (size_t)LD * HD * KTOT;
    if (idx >= total) return;
    int l   = (int)(idx / ((size_t)HD * KTOT));
    int rem = (int)(idx % ((size_t)HD * KTOT));
    int o = rem / KTOT;
    int k = rem % KTOT;
    float v = (k < KREL)
            ? W_rel[((size_t)l * KREL + k) * HD + o]
            : W_self[((size_t)l * HD + (k - KREL)) * HD + o];
    WT[idx] = v;
}

// ---------------------------------------------------------------------------
// 1) Edge scatter:  agg[(dst*R + etype), c] += h[src, c]
//    thread-per-(edge, 4 channels): one wave == one edge, uniform index loads,
//    one coalesced 512B float4 row gather, 4 f32 atomics per thread (L2-resident)
// ---------------------------------------------------------------------------
__global__ void rgcn_scatter(const float* __restrict__ hin, int hin_stride,
                             const int* __restrict__ src,
                             const int* __restrict__ dst,
                             const int* __restrict__ etype,
                             float* __restrict__ agg) {
    size_t idx = (size_t)blockIdx.x * blockDim.x + threadIdx.x;
    size_t total = (size_t)ED * 32;
    if (idx >= total) return;
    int e = (int)(idx >> 5);
    int q = (int)(idx & 31) << 2;          // channel base 0..124
    const float4 v = *(const float4*)(hin + (size_t)src[e] * hin_stride + q);
    float* p = agg + ((size_t)dst[e] * RD + etype[e]) * HD + q;
    atomicAdd(p + 0, v.x);
    atomicAdd(p + 1, v.y);
    atomicAdd(p + 2, v.z);
    atomicAdd(p + 3, v.w);
}

// ---------------------------------------------------------------------------
// 2) Fused RGCN layer GEMM with fp32 WMMA:
//    out[n, o] = relu( agg[n,:] @ Wrel + h[n,:] @ Wself )   via WT[o][k]
//    Block = 256 threads = 8 waves. blockIdx.x = 16-row tile; wave = 16-col tile.
//    WMMA f32 16x16x4 layouts per CDNA5 ISA:
//      A: lane<16 -> M=lane, K={kk,kk+1}; lane>=16 -> M=lane-16, K={kk+2,kk+3}
//      B: lane<16 -> N=lane, K={kk,kk+1}; lane>=16 -> N=lane-16, K={kk+2,kk+3}
//      C/D vgpr i: lanes0-15 M=i, lanes16-31 M=i+8, N=lane&15
// ---------------------------------------------------------------------------
__global__ void rgcn_gemm(const float* __restrict__ hin, int hin_stride,
                          const float* __restrict__ agg,   // [N, 1024]
                          const float* __restrict__ WT,    // [128, 1152] (o-major)
                          float* __restrict__ hout, int hout_stride) {
    const int wave = threadIdx.x >> 5;
    const int lane = threadIdx.x & 31;
    const int lm   = lane & 15;
    const int hi   = lane >> 4;          // which K-pair this lane holds
    const int koff = hi * 2;
    const int m0   = blockIdx.x * 16;
    const int n0   = wave * 16;
    const int m    = m0 + lm;
    const int col  = n0 + lm;

    v8f acc0 = {};
    v8f acc1 = {};

    const float* brow = WT + (size_t)col * KTOT;

    // ---- K part 1: agg @ Wrel  (K = 1024), two interleaved accumulators ----
    const float* arow = agg + (size_t)m * KREL;
    for (int kk = 0; kk < KREL; kk += 8) {
        v2f a0 = *(const v2f*)(arow + kk + koff);
        v2f a1 = *(const v2f*)(arow + kk + 4 + koff);
        v2f b0 = *(const v2f*)(brow + kk + koff);
        v2f b1 = *(const v2f*)(brow + kk + 4 + koff);
        acc0 = __builtin_amdgcn_wmma_f32_16x16x4_f32(false, a0, false, b0,
                                                     (short)0, acc0, false, false);
        acc1 = __builtin_amdgcn_wmma_f32_16x16x4_f32(false, a1, false, b1,
                                                     (short)0, acc1, false, false);
    }

    // ---- K part 2: h @ Wself (K = 128) ----
    const float* hrow = hin + (size_t)m * hin_stride;
    const float* brow2 = brow + KREL;
    for (int kk = 0; kk < HD; kk += 8) {
        v2f a0 = *(const v2f*)(hrow + kk + koff);
        v2f a1 = *(const v2f*)(hrow + kk + 4 + koff);
        v2f b0 = *(const v2f*)(brow2 + kk + koff);
        v2f b1 = *(const v2f*)(brow2 + kk + 4 + koff);
        acc0 = __builtin_amdgcn_wmma_f32_16x16x4_f32(false, a0, false, b0,
                                                     (short)0, acc0, false, false);
        acc1 = __builtin_amdgcn_wmma_f32_16x16x4_f32(false, a1, false, b1,
                                                     (short)0, acc1, false, false);
    }

    // ---- writeback with relu ----
    const int mrow_base = m0 + hi * 8;
#pragma unroll
    for (int i = 0; i < 8; ++i) {
        float v = acc0[i] + acc1[i];
        v = v > 0.0f ? v : 0.0f;
        hout[(size_t)(mrow_base + i) * hout_stride + col] = v;
    }
}

// ---------------------------------------------------------------------------
// 3) Mean-pool accumulation per subgraph
// ---------------------------------------------------------------------------
__global__ void pool_accum(const float* __restrict__ reprs,  // [N, L*H]
                           const int* __restrict__ graph_id,
                           float* __restrict__ g_sum,        // [BS, L*H]
                           float* __restrict__ counts) {     // [BS]
    size_t idx = (size_t)blockIdx.x * blockDim.x + threadIdx.x;
    size_t total = (size_t)NND * (LD * HD);
    if (idx >= total) return;
    int node = (int)(idx / (LD * HD));
    int c    = (int)(idx % (LD * HD));
    int g = graph_id[node];
    atomicAdd(&g_sum[(size_t)g * (LD * HD) + c], reprs[idx]);
    if (c == 0) atomicAdd(&counts[g], 1.0f);
}

// ---------------------------------------------------------------------------
// 4) Tail: all BS=32 math. One block per batch item, 128 threads (t = channel)
// ---------------------------------------------------------------------------
__global__ void tail_kernel(const float* __restrict__ reprs,   // [N, 384]
                            const float* __restrict__ g_sum,   // [BS, 384]
                            const float* __restrict__ counts,  // [BS]
                            const float* __restrict__ proj_W,  // [384, 128]
                            const float* __restrict__ proj_b,  // [128]
                            const float* __restrict__ Zn,      // [64, 128]
                            const float* __restrict__ rel_table, // [R, 128]
                            const float* __restrict__ rep_seq,   // [BS, 128]
                            const float* __restrict__ fc_W,      // [1024]
                            const float* __restrict__ fc_b,      // [1]
                            const int* __restrict__ head_ids,
                            const int* __restrict__ tail_ids,
                            const int* __restrict__ rel_labels,
                            float* __restrict__ out) {           // [BS]
    __shared__ float hf[LD * HD], tf[LD * HD], gm[LD * HD];
    __shared__ float gout[HD], hout[HD], tout[HD], sv[HD];
    __shared__ float p1[KCL], p2[KCL];
    __shared__ float r0[HD], r1[HD], r2[HD], red[HD];
    __shared__ float att[3];

    const int b = blockIdx.x;
    const int t = threadIdx.x;   // 0..127

    const int hid = head_ids[b];
    const int tid = tail_ids[b];
    const float inv_cnt = 1.0f / counts[b];

    for (int k = t; k < LD * HD; k += HD) {
        hf[k] = reprs[(size_t)hid * (LD * HD) + k];
        tf[k] = reprs[(size_t)tid * (LD * HD) + k];
        gm[k] = g_sum[(size_t)b * (LD * HD) + k] * inv_cnt;
    }
    __syncthreads();

    // three projections through proj_W (column t), coalesced across threads
    float accg = 0.f, acch = 0.f, acct = 0.f;
    for (int k = 0; k < LD * HD; ++k) {
        float w = proj_W[(size_t)k * HD + t];
        accg += gm[k] * w;
        acch += hf[k] * w;
        acct += tf[k] * w;
    }
    float pb = proj_b[t];
    accg += pb; acch += pb; acct += pb;
    gout[t] = accg > 0.f ? accg : 0.01f * accg;  // leaky_relu, slope 0.01
    hout[t] = acch;
    tout[t] = acct;
    __syncthreads();

    // cluster logits: threads 0..63, dot over 128 channels
    if (t < KCL) {
        float l1 = 0.f, l2 = 0.f;
        for (int o = 0; o < HD; ++o) {
            float z = Zn[(size_t)t * HD + o];
            l1 += hout[o] * z;
            l2 += tout[o] * z;
        }
        p1[t] = l1; p2[t] = l2;
    }
    __syncthreads();

    if (t == 0) {  // softmax over 64 clusters (tiny; serial is fine)
        float m1 = -1e30f, m2 = -1e30f;
        for (int c = 0; c < KCL; ++c) { m1 = fmaxf(m1, p1[c]); m2 = fmaxf(m2, p2[c]); }
        float s1s = 0.f, s2s = 0.f;
        for (int c = 0; c < KCL; ++c) {
            p1[c] = __expf(p1[c] - m1); s1s += p1[c];
            p2[c] = __expf(p2[c] - m2); s2s += p2[c];
        }
        float i1 = 1.f / s1s, i2 = 1.f / s2s;
        for (int c = 0; c < KCL; ++c) { p1[c] *= i1; p2[c] *= i2; }
    }
    __syncthreads();

    // community summaries: sigmoid(p @ Zn), elementwise product
    float a1 = 0.f, a2 = 0.f;
    for (int c = 0; c < KCL; ++c) {
        float z = Zn[(size_t)c * HD + t];
        a1 += p1[c] * z;
        a2 += p2[c] * z;
    }
    float sg1 = 1.f / (1.f + __expf(-a1));
    float sg2 = 1.f / (1.f + __expf(-a2));
    sv[t] = sg1 * sg2;

    float rel = rel_table[(size_t)rel_labels[b] * HD + t];
    float rs  = rep_seq[(size_t)b * HD + t];
    __syncthreads();

    // view attention: dots of rel_emb with [g_out, rep_seq, s]
    r0[t] = rel * gout[t];
    r1[t] = rel * rs;
    r2[t] = rel * sv[t];
    __syncthreads();
    if (t == 0) {
        float d0 = 0.f, d1 = 0.f, d2 = 0.f;
        for (int i = 0; i < HD; ++i) { d0 += r0[i]; d1 += r1[i]; d2 += r2[i]; }
        float mx = fmaxf(d0, fmaxf(d1, d2));
        float e0 = __expf(d0 - mx), e1 = __expf(d1 - mx), e2 = __expf(d2 - mx);
        float inv = 1.f / (e0 + e1 + e2);
        att[0] = e0 * inv; att[1] = e1 * inv; att[2] = e2 * inv;
    }
    __syncthreads();

    float va = att[0] * gout[t] + att[1] * rs + att[2] * sv[t];

    // final score: [hf(384) | tf(384) | rel(128) | va(128)] . fc_W + fc_b
    float part = 0.f;
    part += hf[t] * fc_W[t] + hf[t + 128] * fc_W[t + 128] + hf[t + 256] * fc_W[t + 256];
    part += tf[t] * fc_W[384 + t] + tf[t + 128] * fc_W[384 + t + 128]
          + tf[t + 256] * fc_W[384 + t + 256];
    part += rel * fc_W[768 + t];
    part += va  * fc_W[896 + t];
    red[t] = part;
    __syncthreads();
    if (t == 0) {
        float s = 0.f;
        for (int i = 0; i < HD; ++i) s += red[i];
        out[b] = s + fc_b[0];
    }
}

// ---------------------------------------------------------------------------
// Launcher
// ---------------------------------------------------------------------------
extern "C" void kernel_launch(void* const* d_in, const int* in_sizes, int n_in,
                              void* d_out, int out_size, void* d_ws, size_t ws_size,
                              hipStream_t stream) {
    const float* x         = (const float*)d_in[0];
    const float* W_rel     = (const float*)d_in[1];   // [L,R,H,H]
    const float* W_self    = (const float*)d_in[2];   // [L,H,H]
    const float* rel_table = (const float*)d_in[3];
    const float* Zn        = (const float*)d_in[4];
    const float* proj_W    = (const float*)d_in[5];
    const float* proj_b    = (const float*)d_in[6];
    const float* fc_W      = (const float*)d_in[7];
    const float* fc_b      = (const float*)d_in[8];
    const float* rep_seq   = (const float*)d_in[9];
    const int*   src       = (const int*)d_in[10];
    const int*   dst       = (const int*)d_in[11];
    const int*   etype     = (const int*)d_in[12];
    const int*   graph_id  = (const int*)d_in[13];
    const int*   head_ids  = (const int*)d_in[14];
    const int*   tail_ids  = (const int*)d_in[15];
    const int*   rel_label = (const int*)d_in[16];
    float* out = (float*)d_out;

    // workspace layout (floats)
    float* reprs  = (float*)d_ws;                         // N * L * H
    float* agg    = reprs + (size_t)NND * LD * HD;        // N * R * H
    float* g_sum  = agg   + (size_t)NND * RD * HD;        // BS * L * H
    float* counts = g_sum + (size_t)BSD * LD * HD;        // BS
    float* WT     = counts + BSD;                         // L * H * KTOT

    hipMemsetAsync(g_sum, 0, ((size_t)BSD * LD * HD + BSD) * sizeof(float), stream);

    const size_t wt_threads = (size_t)LD * HD * KTOT;
    const int wt_blocks = (int)((wt_threads + 255) / 256);
    build_wt<<<wt_blocks, 256, 0, stream>>>(W_rel, W_self, WT);

    const size_t scat_threads = (size_t)ED * 32;
    const int scat_blocks = (int)((scat_threads + 255) / 256);
    const int gemm_blocks = NND / 16;                     // 2000
    const size_t pool_threads = (size_t)NND * LD * HD;
    const int pool_blocks = (int)((pool_threads + 255) / 256);

    for (int l = 0; l < LD; ++l) {
        const float* hin = (l == 0) ? x : (reprs + (size_t)(l - 1) * HD);
        const int hstride = (l == 0) ? HD : (LD * HD);

        hipMemsetAsync(agg, 0, (size_t)NND * RD * HD * sizeof(float), stream);
        rgcn_scatter<<<scat_blocks, 256, 0, stream>>>(hin, hstride, src, dst, etype, agg);
        rgcn_gemm<<<gemm_blocks, 256, 0, stream>>>(
            hin, hstride, agg,
            WT + (size_t)l * HD * KTOT,
            reprs + (size_t)l * HD, LD * HD);
    }

    pool_accum<<<pool_blocks, 256, 0, stream>>>(reprs, graph_id, g_sum, counts);

    tail_kernel<<<BSD, HD, 0, stream>>>(reprs, g_sum, counts, proj_W, proj_b,
                                        Zn, rel_table, rep_seq, fc_W, fc_b,
                                        head_ids, tail_ids, rel_label, out);
}